// GCNNormNodeLabelAggregation_5153960755614
// MI455X (gfx1250) — compile-verified
//
#include <hip/hip_runtime.h>

// GCN norm + SpMM aggregation for MI455X (gfx1250, wave32).
// Memory-bound: ~100MB mandatory HBM traffic; x (25.6MB) is L2-resident
// (192MB L2) so the 410MB of random row gathers are L2 hits. CSR two-phase
// avoids 102M f32 atomics. Aggregation double-buffers neighbor rows through
// LDS with gfx1250 GLOBAL_LOAD_ASYNC_TO_LDS_B128 (ASYNCcnt), overlapping the
// next chunk's DMA with the current chunk's FMAs.

#define THREADS 256
#define CHUNK 16            // neighbors staged per pipeline stage
#define WAVES_PER_BLOCK 8   // 256 threads / wave32

// --- Kernel 1: out[:, :64] = x (float4), zero degree array -----------------
__global__ void k_copy_x_zero_deg(const float* __restrict__ x,
                                  float* __restrict__ out,
                                  int* __restrict__ deg, int n) {
  int t = blockIdx.x * blockDim.x + threadIdx.x;
  int total = n * 16;                       // n rows * 64 floats / 4
  if (t < total) {
    int i = t >> 4;                         // node
    int f4 = t & 15;                        // float4 slot within row
    float4 v = reinterpret_cast<const float4*>(x)[t];
    reinterpret_cast<float4*>(out)[i * 32 + f4] = v;   // out row = 32 float4
  }
  if (t < n) deg[t] = 0;
}

// --- Kernel 2: degree histogram (int atomics into L2) ----------------------
__global__ void k_degree(const int* __restrict__ row,
                         int* __restrict__ deg, int e_cnt) {
  int e = blockIdx.x * blockDim.x + threadIdx.x;
  if (e < e_cnt) {
    __builtin_prefetch(row + e + 8192, 0, 0);   // global_prefetch_b8
    atomicAdd(&deg[row[e]], 1);
  }
}

// --- Kernel 3: dinv = deg > 0 ? rsqrt(deg) : 0 -----------------------------
__global__ void k_dinv(const int* __restrict__ deg,
                       float* __restrict__ dinv, int n) {
  int i = blockIdx.x * blockDim.x + threadIdx.x;
  if (i < n) {
    int d = deg[i];
    dinv[i] = (d > 0) ? rsqrtf((float)d) : 0.0f;
  }
}

// --- Kernel 4: single-workgroup exclusive scan of deg -> row_start, cursor -
__global__ __launch_bounds__(1024) void k_scan(const int* __restrict__ deg,
                                               int* __restrict__ row_start,
                                               int* __restrict__ cursor, int n) {
  __shared__ int lds[1024];
  __shared__ int carry_s;
  int tid = threadIdx.x;
  if (tid == 0) carry_s = 0;
  __syncthreads();
  for (int base = 0; base < n; base += 1024) {
    int i = base + tid;
    int v = (i < n) ? deg[i] : 0;
    lds[tid] = v;
    __syncthreads();
    for (int off = 1; off < 1024; off <<= 1) {   // Hillis-Steele inclusive
      int t = (tid >= off) ? lds[tid - off] : 0;
      __syncthreads();
      lds[tid] += t;
      __syncthreads();
    }
    int incl = lds[tid];
    int carry = carry_s;
    __syncthreads();
    if (i < n) {
      int excl = carry + incl - v;
      row_start[i] = excl;
      cursor[i] = excl;
    }
    if (tid == 1023) carry_s = carry + incl;
    __syncthreads();
  }
  if (tid == 0) row_start[n] = carry_s;
}

// --- Kernel 5: scatter edges into CSR --------------------------------------
__global__ void k_fill(const int* __restrict__ row, const int* __restrict__ col,
                       int* __restrict__ cursor, int* __restrict__ csr_col,
                       int e_cnt) {
  int e = blockIdx.x * blockDim.x + threadIdx.x;
  if (e < e_cnt) {
    __builtin_prefetch(row + e + 8192, 0, 0);
    __builtin_prefetch(col + e + 8192, 0, 0);
    int r = row[e];
    int p = atomicAdd(&cursor[r], 1);
    csr_col[p] = col[e];
  }
}

// --- Kernel 6: one wave32 per node; async double-buffered aggregation ------
// Stage CHUNK neighbor rows (CHUNK x 256B) into wave-private LDS with
// GLOBAL_LOAD_ASYNC_TO_LDS_B128 (2 rows per issue: 32 lanes x 16B), then
// overlap chunk k+1's async DMA with chunk k's FMAs. s_wait_asynccnt 0 at the
// loop head only covers loads issued one full compute phase earlier.
__global__ __launch_bounds__(THREADS) void k_aggregate_async(
    const float* __restrict__ x, const int* __restrict__ row_start,
    const int* __restrict__ csr_col, const float* __restrict__ dinv,
    float* __restrict__ out, int n) {
  __shared__ __align__(16) float smem[2 * WAVES_PER_BLOCK * CHUNK * 64]; // 64KB
  int wslot = threadIdx.x >> 5;
  int lane = threadIdx.x & 31;
  int node = blockIdx.x * WAVES_PER_BLOCK + wslot;
  if (node >= n) return;

  int e0 = row_start[node];
  int e1 = row_start[node + 1];
  float di = dinv[node];
  float2* obuf = reinterpret_cast<float2*>(out + (size_t)node * 128 + 64);

  if (e0 >= e1) {              // isolated node: aggregation is zero
    float2 z; z.x = 0.0f; z.y = 0.0f;
    obuf[lane] = z;
    return;
  }

  float* buf0 = &smem[(size_t)wslot * CHUNK * 64];
  float* buf1 = &smem[(size_t)(WAVES_PER_BLOCK + wslot) * CHUNK * 64];
  unsigned lb0 = (unsigned)(uintptr_t)(void*)buf0;  // LDS byte address
  unsigned lb1 = (unsigned)(uintptr_t)(void*)buf1;
  unsigned long long gbase = (unsigned long long)(uintptr_t)x;

  int half = lane >> 4;                          // which row of the pair
  unsigned boff = (unsigned)(lane & 15) * 16u;   // 16 lanes x 16B = 256B row

  // ---- prologue: stage chunk 0 into buf0 ----
  int e = e0;
  int m = e1 - e; if (m > CHUNK) m = CHUNK;
  int c = 0; float wcur = 0.0f;
  if (lane < m) { c = csr_col[e + lane]; wcur = dinv[c]; }
  for (int p = 0; p < CHUNK / 2; ++p) {
    int jj = p * 2 + half;
    if (jj < m) {
      int cj = __shfl(c, jj, 32);
      unsigned long long g = gbase + (unsigned long long)cj * 256ull + boff;
      unsigned l = lb0 + (unsigned)jj * 256u + boff;
      asm volatile("global_load_async_to_lds_b128 %0, %1, off"
                   :: "v"(l), "v"(g) : "memory");
    }
  }

  int cur = 0;
  float ax = 0.0f, ay = 0.0f;
  for (;;) {
    asm volatile("s_wait_asynccnt 0x0" ::: "memory");  // chunk `cur` resident

    // issue next chunk into the other buffer (overlaps with compute below)
    int enext = e + CHUNK;
    int mn = e1 - enext; if (mn > CHUNK) mn = CHUNK;
    float wnext = 0.0f;
    if (mn > 0) {
      int cn = 0;
      if (lane < mn) { cn = csr_col[enext + lane]; wnext = dinv[cn]; }
      unsigned lbn = (cur == 0) ? lb1 : lb0;
      for (int p = 0; p < CHUNK / 2; ++p) {
        int jj = p * 2 + half;
        if (jj < mn) {
          int cj = __shfl(cn, jj, 32);
          unsigned long long g = gbase + (unsigned long long)cj * 256ull + boff;
          unsigned l = lbn + (unsigned)jj * 256u + boff;
          asm volatile("global_load_async_to_lds_b128 %0, %1, off"
                       :: "v"(l), "v"(g) : "memory");
        }
      }
    }

    // compute on resident chunk: lane owns 2 features (32 lanes x 2 = 64)
    const float* b = (cur == 0) ? buf0 : buf1;
    for (int j = 0; j < m; ++j) {
      float wj = __shfl(wcur, j, 32);
      float2 xv = *reinterpret_cast<const float2*>(b + j * 64 + lane * 2);
      ax = fmaf(wj, xv.x, ax);
      ay = fmaf(wj, xv.y, ay);
    }

    if (mn <= 0) break;
    e = enext; m = mn; wcur = wnext; cur ^= 1;
  }

  float2 r; r.x = di * ax; r.y = di * ay;   // out = dinv[row] * sum(dinv[col]*x[col])
  obuf[lane] = r;
}

extern "C" void kernel_launch(void* const* d_in, const int* in_sizes, int n_in,
                              void* d_out, int out_size, void* d_ws, size_t ws_size,
                              hipStream_t stream) {
  const float* x = (const float*)d_in[0];
  const int* edge = (const int*)d_in[1];   // int32 per harness convention
  int n = in_sizes[0] / 64;                // D_FEAT = 64
  int E = in_sizes[1] / 2;                 // edge_index shape (2, E)
  const int* row = edge;
  const int* col = edge + E;
  float* out = (float*)d_out;

  // Workspace carve-up (all 4-byte aligned): ~8.4 MB total
  char* ws = (char*)d_ws;
  int* deg = (int*)ws;            ws += (size_t)n * sizeof(int);
  int* cursor = (int*)ws;         ws += (size_t)n * sizeof(int);
  int* row_start = (int*)ws;      ws += (size_t)(n + 1) * sizeof(int);
  float* dinv = (float*)ws;       ws += (size_t)n * sizeof(float);
  int* csr_col = (int*)ws;        // E ints

  int copy_blocks = (n * 16 + THREADS - 1) / THREADS;
  int edge_blocks = (E + THREADS - 1) / THREADS;
  int node_blocks = (n + THREADS - 1) / THREADS;
  int agg_blocks = (n + WAVES_PER_BLOCK - 1) / WAVES_PER_BLOCK;

  k_copy_x_zero_deg<<<copy_blocks, THREADS, 0, stream>>>(x, out, deg, n);
  k_degree<<<edge_blocks, THREADS, 0, stream>>>(row, deg, E);
  k_dinv<<<node_blocks, THREADS, 0, stream>>>(deg, dinv, n);
  k_scan<<<1, 1024, 0, stream>>>(deg, row_start, cursor, n);
  k_fill<<<edge_blocks, THREADS, 0, stream>>>(row, col, cursor, csr_col, E);
  k_aggregate_async<<<agg_blocks, THREADS, 0, stream>>>(x, row_start, csr_col,
                                                        dinv, out, n);
}